// myAttentionalPropagation_64415919505665
// MI455X (gfx1250) — compile-verified
//
#include <hip/hip_runtime.h>

// Problem constants (from the reference)
#define BB 2
#define NN 20000
#define FF 128
#define CC 128
#define EE 640000
#define EP (EE + NN)   // edges + self loops

typedef float v2f __attribute__((ext_vector_type(2)));
typedef float v8f __attribute__((ext_vector_type(8)));

// Monotone float <-> uint encoding so we can use integer atomicMax for the
// per-destination segment max (every dst has a self-loop, so it is always set).
__device__ __forceinline__ unsigned fenc(float f) {
  unsigned b = __float_as_uint(f);
  return (b & 0x80000000u) ? ~b : (b | 0x80000000u);
}
__device__ __forceinline__ float fdec(unsigned k) {
  unsigned b = (k & 0x80000000u) ? (k & 0x7FFFFFFFu) : ~k;
  return __uint_as_float(b);
}
__device__ __forceinline__ float lrelu(float z) { return z > 0.f ? z : 0.2f * z; }

// Non-returning, relaxed, agent-scope float atomic add -> should lower to the
// native GLOBAL_ATOMIC_ADD_F32 (no CAS retry loop).
__device__ __forceinline__ void fadd_agent(float* p, float v) {
  (void)__hip_atomic_fetch_add(p, v, __ATOMIC_RELAXED, __HIP_MEMORY_SCOPE_AGENT);
}

__device__ __forceinline__ void edge_sd(const long long* __restrict__ ei,
                                        int e, int& s, int& d) {
  if (e < EE) { s = (int)ei[e]; d = (int)ei[EE + e]; }
  else        { s = e - EE; d = e - EE; }
}

// ---------------------------------------------------------------------------
// Init: out = bias (folds the final +bias), segmax = -inf encoding, denom = 0.
// ---------------------------------------------------------------------------
__global__ void gat_init(float* __restrict__ out, const float* __restrict__ bias,
                         unsigned* __restrict__ segmax, float* __restrict__ denom) {
  size_t i = (size_t)blockIdx.x * blockDim.x + threadIdx.x;
  if (i < (size_t)BB * NN * CC) out[i] = bias[i & (CC - 1)];
  if (i < (size_t)BB * NN) { segmax[i] = 0u; denom[i] = 0.f; }
}

// ---------------------------------------------------------------------------
// h = x @ W^T via native f32 WMMA (16x16x4). One wave per 16-row tile.
// A (16x4): lanes 0-15 hold M=0..15 {K=k0,k0+1}; lanes 16-31 hold {K=k0+2,k0+3}.
// B (4x16): lane n-col in [0,15]; VGPR0={K=k0 | k0+2}, VGPR1={K=k0+1 | k0+3}.
// D (16x16): lane col = lane&15, VGPR v -> row M = v + (lane>=16 ? 8 : 0).
// ---------------------------------------------------------------------------
__global__ __launch_bounds__(32)
void gat_gemm_wmma(const float* __restrict__ x, const float* __restrict__ W,
                   float* __restrict__ h) {
  const int lane = threadIdx.x;       // 0..31 (wave32)
  const int tile = blockIdx.x;        // 0..NN/16-1
  const int b    = blockIdx.y;
  const int r    = lane & 15;
  const int hi   = lane >> 4;         // K-pair selector
  const float* xr = x + ((size_t)b * NN + (size_t)tile * 16 + r) * FF;

  for (int ct = 0; ct < CC / 16; ++ct) {
    v8f acc = {0.f, 0.f, 0.f, 0.f, 0.f, 0.f, 0.f, 0.f};
    const float* wc = W + (size_t)(ct * 16 + r) * FF; // W[c, :], c = output col
    for (int k0 = 0; k0 < FF; k0 += 4) {
      const int ka = k0 + 2 * hi;
      v2f a;  a.x  = xr[ka]; a.y  = xr[ka + 1];
      v2f bm; bm.x = wc[ka]; bm.y = wc[ka + 1];
      acc = __builtin_amdgcn_wmma_f32_16x16x4_f32(
          /*neg_a=*/false, a, /*neg_b=*/false, bm,
          /*c_mod=*/(short)0, acc, /*reuse_a=*/false, /*reuse_b=*/false);
    }
    float* hb = h + ((size_t)b * NN + (size_t)tile * 16 + hi * 8) * CC
                  + ct * 16 + r;
#pragma unroll
    for (int v = 0; v < 8; ++v) hb[(size_t)v * CC] = acc[v];
  }
}

// ---------------------------------------------------------------------------
// Per-node attention scalars: a_src[n] = h[n].att_src, a_dst[n] = h[n].att_dst
// One wave per node; 4 channels per lane; shuffle reduction across 32 lanes.
// ---------------------------------------------------------------------------
__global__ __launch_bounds__(256)
void gat_attn(const float* __restrict__ h, const float* __restrict__ att_src,
              const float* __restrict__ att_dst, float* __restrict__ a_src,
              float* __restrict__ a_dst) {
  const int lane = threadIdx.x & 31;
  const int gw = (int)((blockIdx.x * blockDim.x + threadIdx.x) >> 5);
  if (gw >= BB * NN) return;
  const float4 hv = *(const float4*)(h + (size_t)gw * CC + lane * 4);
  const float4 as = *(const float4*)(att_src + lane * 4);
  const float4 ad = *(const float4*)(att_dst + lane * 4);
  float ps = hv.x * as.x + hv.y * as.y + hv.z * as.z + hv.w * as.w;
  float pd = hv.x * ad.x + hv.y * ad.y + hv.z * ad.z + hv.w * ad.w;
#pragma unroll
  for (int off = 16; off; off >>= 1) {
    ps += __shfl_xor(ps, off, 32);
    pd += __shfl_xor(pd, off, 32);
  }
  if (lane == 0) { a_src[gw] = ps; a_dst[gw] = pd; }
}

// ---------------------------------------------------------------------------
// Edge pass 1: segment max of leaky-relu logits (ordered-uint atomicMax).
// ---------------------------------------------------------------------------
__global__ __launch_bounds__(256)
void gat_edge_max(const long long* __restrict__ ei, const float* __restrict__ a_src,
                  const float* __restrict__ a_dst, unsigned* __restrict__ segmax) {
  const int idx = blockIdx.x * blockDim.x + threadIdx.x;
  if (idx >= BB * EP) return;
  const int b = idx / EP, e = idx - b * EP;
  int s, d; edge_sd(ei, e, s, d);
  const float lg = lrelu(a_src[b * NN + s] + a_dst[b * NN + d]);
  atomicMax(&segmax[b * NN + d], fenc(lg));
}

// ---------------------------------------------------------------------------
// Edge pass 2: denom[d] += exp(logit - segmax[d])
// ---------------------------------------------------------------------------
__global__ __launch_bounds__(256)
void gat_edge_sum(const long long* __restrict__ ei, const float* __restrict__ a_src,
                  const float* __restrict__ a_dst, const unsigned* __restrict__ segmax,
                  float* __restrict__ denom) {
  const int idx = blockIdx.x * blockDim.x + threadIdx.x;
  if (idx >= BB * EP) return;
  const int b = idx / EP, e = idx - b * EP;
  int s, d; edge_sd(ei, e, s, d);
  const float lg = lrelu(a_src[b * NN + s] + a_dst[b * NN + d]);
  const float ex = __expf(lg - fdec(segmax[b * NN + d]));
  fadd_agent(&denom[b * NN + d], ex);
}

// ---------------------------------------------------------------------------
// Edge pass 3: out[d] += alpha * h[s]. One wave per edge; float4 per lane;
// native f32 global atomic adds (working set is L2-resident: ~20MB << 192MB L2).
// ---------------------------------------------------------------------------
__global__ __launch_bounds__(256)
void gat_edge_scatter(const long long* __restrict__ ei, const float* __restrict__ a_src,
                      const float* __restrict__ a_dst, const unsigned* __restrict__ segmax,
                      const float* __restrict__ denom, const float* __restrict__ h,
                      float* __restrict__ out) {
  const int lane = threadIdx.x & 31;
  const int wid = (int)((blockIdx.x * blockDim.x + threadIdx.x) >> 5);
  const int nw = (int)((gridDim.x * blockDim.x) >> 5);
  for (int idx = wid; idx < BB * EP; idx += nw) {
    const int b = idx / EP, e = idx - b * EP;
    int s, d; edge_sd(ei, e, s, d);
    const float lg = lrelu(a_src[b * NN + s] + a_dst[b * NN + d]);
    const float alpha = __expf(lg - fdec(segmax[b * NN + d])) / denom[b * NN + d];
    const float4 hv = *(const float4*)(h + ((size_t)b * NN + s) * CC + lane * 4);
    float* o = out + ((size_t)b * NN + d) * CC + lane * 4;
    fadd_agent(o + 0, alpha * hv.x);
    fadd_agent(o + 1, alpha * hv.y);
    fadd_agent(o + 2, alpha * hv.z);
    fadd_agent(o + 3, alpha * hv.w);
  }
}

extern "C" void kernel_launch(void* const* d_in, const int* in_sizes, int n_in,
                              void* d_out, int out_size, void* d_ws, size_t ws_size,
                              hipStream_t stream) {
  const float*     x       = (const float*)d_in[0];      // [B,N,F]
  const long long* ei      = (const long long*)d_in[1];  // [2,E] int64
  const float*     W       = (const float*)d_in[2];      // [C,F]
  const float*     att_src = (const float*)d_in[3];      // [1,C]
  const float*     att_dst = (const float*)d_in[4];      // [1,C]
  const float*     bias    = (const float*)d_in[5];      // [C]
  float* out = (float*)d_out;                            // [B,N,C]

  // Workspace layout
  float*    h      = (float*)d_ws;                   // B*N*C
  float*    a_src  = h + (size_t)BB * NN * CC;       // B*N
  float*    a_dst  = a_src + (size_t)BB * NN;        // B*N
  unsigned* segmax = (unsigned*)(a_dst + (size_t)BB * NN);
  float*    denom  = (float*)(segmax + (size_t)BB * NN);

  const int totOut  = BB * NN * CC;                  // 5,120,000
  const int totEdge = BB * EP;                       // 1,320,000

  gat_init<<<(totOut + 255) / 256, 256, 0, stream>>>(out, bias, segmax, denom);

  gat_gemm_wmma<<<dim3(NN / 16, BB), 32, 0, stream>>>(x, W, h);

  gat_attn<<<(BB * NN * 32 + 255) / 256, 256, 0, stream>>>(h, att_src, att_dst,
                                                           a_src, a_dst);

  gat_edge_max<<<(totEdge + 255) / 256, 256, 0, stream>>>(ei, a_src, a_dst, segmax);

  gat_edge_sum<<<(totEdge + 255) / 256, 256, 0, stream>>>(ei, a_src, a_dst,
                                                          segmax, denom);

  gat_edge_scatter<<<4096, 256, 0, stream>>>(ei, a_src, a_dst, segmax, denom,
                                             h, out);
}